// RelationalContextEncoder_29557964931729
// MI455X (gfx1250) — compile-verified
//
#include <hip/hip_runtime.h>

typedef __attribute__((ext_vector_type(2))) float v2f;
typedef __attribute__((ext_vector_type(8))) float v8f;

#define NUM_ENT 7
#define DIM_ENT 4
#define HDIM 512
#define ROWS 42        // 7 * 6 valid pair rows
#define ROWS_PAD 48    // 3 M-tiles of 16
#define HPAD 516       // h row stride (floats); 516 % 64 == 4 -> conflict-free A loads

__global__ __launch_bounds__(256) void rce_fused_kernel(
    const float* __restrict__ ctx, const float* __restrict__ W1, const float* __restrict__ b1,
    const float* __restrict__ W2, const float* __restrict__ b2,
    const float* __restrict__ Wp, const float* __restrict__ bp,
    float* __restrict__ out)
{
    __shared__ float s_h[ROWS_PAD * HPAD];      // 99072 B : layer-1 activations
    __shared__ float s_feats[ROWS_PAD * 12];    //  2304 B : 9 feats padded to 12
    __shared__ float s_pool[NUM_ENT * HDIM];    // 14336 B : j-pool accumulators

    const int b = blockIdx.x;
    const int t = threadIdx.x;
    const float* cb = ctx + b * (NUM_ENT * DIM_ENT);

    // ---- zero pool accumulators ----
    for (int idx = t; idx < NUM_ENT * HDIM; idx += 256) s_pool[idx] = 0.0f;

    // ---- pair features ----
    if (t < ROWS_PAD) {
        float f[9];
        if (t < ROWS) {
            const int i  = t / 6, jp = t % 6;
            const int j  = jp + (jp >= i ? 1 : 0);   // off-diagonal index
            const float ei0 = cb[i*4+0], ei1 = cb[i*4+1], ei2 = cb[i*4+2], ei3 = cb[i*4+3];
            const float ej0 = cb[j*4+0], ej1 = cb[j*4+1], ej2 = cb[j*4+2], ej3 = cb[j*4+3];
            const float d0 = ei0 - ej0, d1 = ei1 - ej1, d2 = ei2 - ej2, d3 = ei3 - ej3;
            const float dist = sqrtf(d0*d0 + d1*d1);
            f[0]=ei2; f[1]=ei3; f[2]=ej2; f[3]=ej3;
            f[4]=d0;  f[5]=d1;  f[6]=d2;  f[7]=d3;  f[8]=dist;
        } else {
            #pragma unroll
            for (int q = 0; q < 9; ++q) f[q] = 0.0f;
        }
        #pragma unroll
        for (int q = 0; q < 9; ++q) s_feats[t*12 + q] = f[q];
    }

    // ---- prop head: out[b, i, 0:512] = relu(app @ Wp + bp) ----
    for (int idx = t; idx < NUM_ENT * HDIM; idx += 256) {
        const int i = idx >> 9, n = idx & (HDIM - 1);
        const float a0 = cb[i*4+2], a1 = cb[i*4+3];
        float v = fmaf(a0, Wp[n], fmaf(a1, Wp[HDIM + n], bp[n]));
        out[(size_t)(b*NUM_ENT + i)*1024 + n] = fmaxf(v, 0.0f);
    }

    __syncthreads();

    // ---- layer 1: h = relu(feats @ W1 + b1), K=9 via plain FMA ----
    for (int idx = t; idx < ROWS_PAD * HDIM; idx += 256) {
        const int p = idx >> 9, n = idx & (HDIM - 1);
        float v = 0.0f;
        if (p < ROWS) {
            v = b1[n];
            #pragma unroll
            for (int q = 0; q < 9; ++q) v = fmaf(s_feats[p*12 + q], W1[q*HDIM + n], v);
            v = fmaxf(v, 0.0f);
        }
        s_h[p*HPAD + n] = v;   // padding rows hold exact zeros
    }

    __syncthreads();

    // ---- layer 2: rel = relu(h @ W2 + b2) via V_WMMA_F32_16X16X4_F32 ----
    const int wave = t >> 5;
    const int lane = t & 31;
    const int lo   = lane & 15;   // A: row M offset / B: col N offset
    const int hi   = lane >> 4;   // K sub-phase select (lanes 16-31 -> K+2)

    for (int nti = 0; nti < 4; ++nti) {
        const int nt   = wave * 4 + nti;       // 8 waves x 4 = all 32 N-tiles
        const int ncol = nt * 16 + lo;
        v8f acc0 = {}; v8f acc1 = {}; v8f acc2 = {};

        for (int k = 0; k < HDIM; k += 4) {
            const int kk = k + 2*hi;
            // B fragment (4x16 of W2), reused across the 3 M-tiles
            v2f bm;
            bm.x = W2[(size_t)kk      * HDIM + ncol];
            bm.y = W2[(size_t)(kk+1)  * HDIM + ncol];
            // A fragments (16x4 of h) from LDS, conflict-free (stride 516)
            v2f a0, a1, a2;
            a0.x = s_h[( 0 + lo)*HPAD + kk];  a0.y = s_h[( 0 + lo)*HPAD + kk + 1];
            a1.x = s_h[(16 + lo)*HPAD + kk];  a1.y = s_h[(16 + lo)*HPAD + kk + 1];
            a2.x = s_h[(32 + lo)*HPAD + kk];  a2.y = s_h[(32 + lo)*HPAD + kk + 1];

            acc0 = __builtin_amdgcn_wmma_f32_16x16x4_f32(false, a0, false, bm, (short)0, acc0, false, false);
            acc1 = __builtin_amdgcn_wmma_f32_16x16x4_f32(false, a1, false, bm, (short)0, acc1, false, false);
            acc2 = __builtin_amdgcn_wmma_f32_16x16x4_f32(false, a2, false, bm, (short)0, acc2, false, false);
        }

        // bias + relu + mean-pool accumulate (D layout: VGPR r -> M = r + 8*hi)
        const float bias = b2[ncol];
        #pragma unroll
        for (int r = 0; r < 8; ++r) {
            const int m0 =  0 + r + 8*hi;
            const int m1 = 16 + r + 8*hi;
            const int m2 = 32 + r + 8*hi;
            const float v0 = fmaxf(acc0[r] + bias, 0.0f);
            const float v1 = fmaxf(acc1[r] + bias, 0.0f);
            const float v2 = fmaxf(acc2[r] + bias, 0.0f);
            atomicAdd(&s_pool[(m0/6)*HDIM + ncol], v0);          // m0,m1 < 42 always
            atomicAdd(&s_pool[(m1/6)*HDIM + ncol], v1);
            if (m2 < ROWS) atomicAdd(&s_pool[(m2/6)*HDIM + ncol], v2);
        }
    }

    __syncthreads();

    // ---- write pooled mean: out[b, i, 512:1024] ----
    const float inv6 = 1.0f / 6.0f;
    for (int idx = t; idx < NUM_ENT * HDIM; idx += 256) {
        const int i = idx >> 9, n = idx & (HDIM - 1);
        out[(size_t)(b*NUM_ENT + i)*1024 + 512 + n] = s_pool[idx] * inv6;
    }
}

extern "C" void kernel_launch(void* const* d_in, const int* in_sizes, int n_in,
                              void* d_out, int out_size, void* d_ws, size_t ws_size,
                              hipStream_t stream) {
    const float* ctx = (const float*)d_in[0];
    const float* W1  = (const float*)d_in[1];
    const float* b1  = (const float*)d_in[2];
    const float* W2  = (const float*)d_in[3];
    const float* b2  = (const float*)d_in[4];
    const float* Wp  = (const float*)d_in[5];
    const float* bp  = (const float*)d_in[6];
    float* out = (float*)d_out;

    const int B = in_sizes[0] / (NUM_ENT * DIM_ENT);   // 8192
    rce_fused_kernel<<<B, 256, 0, stream>>>(ctx, W1, b1, W2, b2, Wp, bp, out);
}